// Encoder_4252017623215
// MI455X (gfx1250) — compile-verified
//
#include <hip/hip_runtime.h>

// ---------------------------------------------------------------------------
// CDNA5 (gfx1250, wave32) fused MLP encoder.
//   nodes: [N,31]->128->128->128->LayerNorm   (V_WMMA_F32_16X16X4_F32)
//   edges: [E, 4]->128->128->128->LayerNorm
// All weights resident in LDS (CDNA5: 320KB/WGP), pair-interleaved so each
// WMMA B fragment is a single conflict-free ds_load_b64. 8 waves/WG, each
// wave computes a 16-row tile fully in registers (8x v8f accumulators),
// A fragments preloaded once per layer.
// ---------------------------------------------------------------------------

typedef __attribute__((ext_vector_type(2))) float v2f;
typedef __attribute__((ext_vector_type(8))) float v8f;

#define HDIM    128
#define PSTRIDE 288   // floats per weight pair-row: 128 pairs (256) + 32 pad
                      // -> pair rows p and p+1 hit disjoint 32-bank halves
#define APAD    132   // LDS row stride for activations (16 lanes -> 16 banks)
#define NWAVES  8
#define TILE_M  16

// D(16x128) = A(16x(4*NK)) @ W; A in LDS (stride APAD), W pair-interleaved:
// W[p*PSTRIDE + 2*n + r] = Worig[2p + r][n].  acc[t] = cols 16t..16t+15.
template <int NK>
__device__ __forceinline__ void gemm_tile(const float* __restrict__ A,
                                          const float* __restrict__ Wp,
                                          int half, int l16, v8f acc[8]) {
  v2f a[NK];
#pragma unroll
  for (int ks = 0; ks < NK; ++ks)   // lanes 16-31 carry K+2,K+3 per ISA layout
    a[ks] = *(const v2f*)(A + l16 * APAD + 4 * ks + 2 * half);
#pragma unroll
  for (int t = 0; t < 8; ++t) {
    v8f c = {0.f, 0.f, 0.f, 0.f, 0.f, 0.f, 0.f, 0.f};
#pragma unroll
    for (int ks = 0; ks < NK; ++ks) {
      const int p = 2 * ks + half;  // weight pair-row for this half
      v2f b = *(const v2f*)(Wp + p * PSTRIDE + 2 * (16 * t + l16));
      c = __builtin_amdgcn_wmma_f32_16x16x4_f32(
          /*neg_a=*/false, a[ks], /*neg_b=*/false, b,
          /*c_mod=*/(short)0, c, /*reuse_a=*/false, /*reuse_b=*/false);
    }
    acc[t] = c;
  }
}

// Global [K][128] row-major -> LDS pair-interleaved, zero-padding rows >= Kvalid.
__device__ __forceinline__ void fill_w(float* __restrict__ dst,
                                       const float* __restrict__ src,
                                       int Kvalid, int Kpad, int tid) {
  for (int i = tid; i < Kpad * HDIM; i += 256) {
    const int k = i >> 7, n = i & 127;
    const float v = (k < Kvalid) ? src[k * HDIM + n] : 0.0f;
    dst[(k >> 1) * PSTRIDE + 2 * n + (k & 1)] = v;
  }
}

template <int K0PAD, bool IS_EDGE>
__global__ __launch_bounds__(256, 1) void encoder_mlp(
    const int* __restrict__ materials, const float* __restrict__ velocities,
    const float* __restrict__ positions, const int* __restrict__ nbr,
    const float* __restrict__ matW, const float* __restrict__ matB,
    const float* __restrict__ w0, const float* __restrict__ b0,
    const float* __restrict__ w1, const float* __restrict__ b1,
    const float* __restrict__ w2, const float* __restrict__ b2,
    const float* __restrict__ lng, const float* __restrict__ lnb,
    float* __restrict__ out, int rows, int k0_valid) {
  extern __shared__ float smem[];
  float* sW0 = smem;                            // (K0PAD/2) x PSTRIDE
  float* sW1 = sW0 + (K0PAD / 2) * PSTRIDE;     // 64 x PSTRIDE
  float* sW2 = sW1 + (HDIM / 2) * PSTRIDE;      // 64 x PSTRIDE
  float* sB0 = sW2 + (HDIM / 2) * PSTRIDE;
  float* sB1 = sB0 + HDIM;
  float* sB2 = sB1 + HDIM;
  float* sG  = sB2 + HDIM;
  float* sBt = sG + HDIM;
  float* sAall = sBt + HDIM;                    // NWAVES x 16 x APAD

  const int tid = threadIdx.x;

  fill_w(sW0, w0, k0_valid, K0PAD, tid);
  fill_w(sW1, w1, HDIM, HDIM, tid);
  fill_w(sW2, w2, HDIM, HDIM, tid);
  for (int i = tid; i < HDIM; i += 256) {
    sB0[i] = b0[i]; sB1[i] = b1[i]; sB2[i] = b2[i];
    sG[i] = lng[i]; sBt[i] = lnb[i];
  }
  __syncthreads();

  const int lane = tid & 31;
  const int wave = tid >> 5;
  const int half = lane >> 4;   // which 16-lane half
  const int l16  = lane & 15;
  float* A = sAall + wave * TILE_M * APAD;      // wave-private activation tile

  const int tiles  = rows >> 4;                 // N, E are multiples of 16
  const int gw     = blockIdx.x * NWAVES + wave;
  const int stride = gridDim.x * NWAVES;

  for (int tile = gw; tile < tiles; tile += stride) {
    const int r0 = tile << 4;

    // ---- build 16-row feature tile into A ----
    if (IS_EDGE) {
      for (int i = lane; i < TILE_M * 4; i += 32) {
        const int r = i >> 2, c = i & 3;
        const int e = r0 + r;
        const int i0 = nbr[2 * e], i1 = nbr[2 * e + 1];
        const float dx = positions[3 * i0 + 0] - positions[3 * i1 + 0];
        const float dy = positions[3 * i0 + 1] - positions[3 * i1 + 1];
        const float dz = positions[3 * i0 + 2] - positions[3 * i1 + 2];
        const float v = (c == 0) ? dx : (c == 1) ? dy : (c == 2) ? dz
                        : sqrtf(dx * dx + dy * dy + dz * dz);
        A[r * APAD + c] = v;
      }
    } else {
      for (int i = lane; i < TILE_M * 32; i += 32) {
        const int r = i >> 5, c = i & 31;
        float v;
        if (c < 15)      v = velocities[(r0 + r) * 15 + c];
        else if (c < 31) v = matW[materials[r0 + r] * 16 + (c - 15)] + matB[c - 15];
        else             v = 0.0f;  // K padding
        A[r * APAD + c] = v;
      }
    }

    v8f acc[8];

    // ---- layer 0: K = K0PAD, bias + relu back to A ----
    gemm_tile<K0PAD / 4>(A, sW0, half, l16, acc);
#pragma unroll
    for (int t = 0; t < 8; ++t) {
      const float bv = sB0[16 * t + l16];
#pragma unroll
      for (int j = 0; j < 8; ++j)
        A[(j + 8 * half) * APAD + 16 * t + l16] = fmaxf(acc[t][j] + bv, 0.0f);
    }

    // ---- layer 1: K = 128, bias + relu back to A ----
    gemm_tile<HDIM / 4>(A, sW1, half, l16, acc);
#pragma unroll
    for (int t = 0; t < 8; ++t) {
      const float bv = sB1[16 * t + l16];
#pragma unroll
      for (int j = 0; j < 8; ++j)
        A[(j + 8 * half) * APAD + 16 * t + l16] = fmaxf(acc[t][j] + bv, 0.0f);
    }

    // ---- layer 2: K = 128, bias only, then fused LayerNorm + store ----
    gemm_tile<HDIM / 4>(A, sW2, half, l16, acc);
#pragma unroll
    for (int t = 0; t < 8; ++t) {
      const float bv = sB2[16 * t + l16];
#pragma unroll
      for (int j = 0; j < 8; ++j) acc[t][j] += bv;
    }

#pragma unroll
    for (int j = 0; j < 8; ++j) {
      // row (j + 8*half) lives across the 16 lanes of this half, 8 col-tiles
      float s = 0.f, q = 0.f;
#pragma unroll
      for (int t = 0; t < 8; ++t) { const float v = acc[t][j]; s += v; q += v * v; }
#pragma unroll
      for (int off = 1; off < 16; off <<= 1) {   // stays within the half
        s += __shfl_xor(s, off, 32);
        q += __shfl_xor(q, off, 32);
      }
      const float mu  = s * (1.0f / 128.0f);
      const float var = q * (1.0f / 128.0f) - mu * mu;
      const float inv = rsqrtf(var + 1e-5f);
      const size_t row = (size_t)(r0 + j + 8 * half);
#pragma unroll
      for (int t = 0; t < 8; ++t) {
        const int col = 16 * t + l16;
        out[row * HDIM + col] = (acc[t][j] - mu) * inv * sG[col] + sBt[col];
      }
    }
  }
}

__global__ void copy_idx_kernel(const int* __restrict__ src,
                                int* __restrict__ dst, int n) {
  const int i = blockIdx.x * blockDim.x + threadIdx.x;
  if (i < n) dst[i] = src[i];
}

extern "C" void kernel_launch(void* const* d_in, const int* in_sizes, int n_in,
                              void* d_out, int out_size, void* d_ws,
                              size_t ws_size, hipStream_t stream) {
  (void)n_in; (void)out_size; (void)d_ws; (void)ws_size;

  const int*   materials  = (const int*)d_in[0];
  const float* velocities = (const float*)d_in[1];
  const float* positions  = (const float*)d_in[2];
  const int*   nbr        = (const int*)d_in[3];
  const float* matW = (const float*)d_in[4];
  const float* matB = (const float*)d_in[5];
  const float* nw0 = (const float*)d_in[6];
  const float* nb0 = (const float*)d_in[7];
  const float* nw1 = (const float*)d_in[8];
  const float* nb1 = (const float*)d_in[9];
  const float* nw2 = (const float*)d_in[10];
  const float* nb2 = (const float*)d_in[11];
  const float* nlng = (const float*)d_in[12];
  const float* nlnb = (const float*)d_in[13];
  const float* ew0 = (const float*)d_in[14];
  const float* eb0 = (const float*)d_in[15];
  const float* ew1 = (const float*)d_in[16];
  const float* eb1 = (const float*)d_in[17];
  const float* ew2 = (const float*)d_in[18];
  const float* eb2 = (const float*)d_in[19];
  const float* elng = (const float*)d_in[20];
  const float* elnb = (const float*)d_in[21];

  const int N = in_sizes[0];
  const int E = in_sizes[3] / 2;

  float* out_nodes = (float*)d_out;
  float* out_edges = out_nodes + (size_t)N * HDIM;
  int*   out_idx   = (int*)(out_edges + (size_t)E * HDIM);

  const size_t lds_node =
      (size_t)((16 + 64 + 64) * PSTRIDE + 5 * HDIM + NWAVES * TILE_M * APAD) *
      sizeof(float);  // ~230.5 KB
  const size_t lds_edge =
      (size_t)((2 + 64 + 64) * PSTRIDE + 5 * HDIM + NWAVES * TILE_M * APAD) *
      sizeof(float);  // ~214.8 KB

  const int node_tiles = N / 16;                          // 3125
  const int edge_tiles = E / 16;                          // 50000
  int node_wgs = (node_tiles + NWAVES - 1) / NWAVES;      // 1 tile per wave
  int edge_wgs = edge_tiles / 32;                         // ~4 tiles per wave
  if (edge_wgs > 1536) edge_wgs = 1536;
  if (edge_wgs < 1) edge_wgs = 1;
  if (node_wgs < 1) node_wgs = 1;

  encoder_mlp<32, false><<<node_wgs, 256, lds_node, stream>>>(
      materials, velocities, nullptr, nullptr, matW, matB,
      nw0, nb0, nw1, nb1, nw2, nb2, nlng, nlnb, out_nodes, N, 31);

  encoder_mlp<4, true><<<edge_wgs, 256, lds_edge, stream>>>(
      nullptr, nullptr, positions, nbr, nullptr, nullptr,
      ew0, eb0, ew1, eb1, ew2, eb2, elng, elnb, out_edges, E, 4);

  const int nidx = 2 * E;
  copy_idx_kernel<<<(nidx + 255) / 256, 256, 0, stream>>>(nbr, out_idx, nidx);
}